// GCN_18837726560519
// MI455X (gfx1250) — compile-verified
//
#include <hip/hip_runtime.h>
#include <cstdint>
#include <cstddef>

// GCN 2-layer inference for MI455X (gfx1250, wave32).
//   N=100000 nodes, E=1.6M edges, F_IN=256, H=64, C=32
// GEMMs use V_WMMA_F32_16X16X4_F32 (exact fp32 matrix pipe).
// Edge aggregation uses hardware global_atomic_add_f32; h/agg buffers
// (25.6MB) fit in the 192MB L2, keeping scatter traffic on-chip.

typedef float v2f __attribute__((ext_vector_type(2)));
typedef float v8f __attribute__((ext_vector_type(8)));

constexpr int KF = 256;   // F_IN
constexpr int HD = 64;    // hidden
constexpr int CD = 32;    // classes

// ----------------- degree / symmetric norm -----------------
__global__ void k_deg_init(float* __restrict__ deg, int n) {
  int i = blockIdx.x * blockDim.x + threadIdx.x;
  if (i < n) deg[i] = 1.0f;                    // self-loop weight
}

__global__ void k_deg_edges(const int* __restrict__ dst, const float* __restrict__ w,
                            float* __restrict__ deg, int e) {
  int i = blockIdx.x * blockDim.x + threadIdx.x;
  if (i < e) unsafeAtomicAdd(&deg[dst[i]], w[i]);   // global_atomic_add_f32
}

__global__ void k_dinv(float* __restrict__ deg, int n) {
  int i = blockIdx.x * blockDim.x + threadIdx.x;
  if (i < n) { float d = deg[i]; deg[i] = d > 0.f ? rsqrtf(d) : 0.f; }
}

// ----------------- WMMA f32 GEMM:  C[M,NOUT] = A[M,K] * B[K,NOUT] -----------------
// One wave computes one 16x16 output tile, accumulating 16x16x4 f32 WMMAs over K.
// A fragment (16x4 f32, 2 VGPRs): lanes 0-15 hold rows M=0..15 with (K=k, K=k+1),
//                                 lanes 16-31 hold same rows with (K=k+2, K=k+3).
// B fragment (4x16 f32, 2 VGPRs): lanes 0-15 hold cols N=0..15 of rows (k, k+1),
//                                 lanes 16-31 of rows (k+2, k+3).
// C/D (16x16 f32, 8 VGPRs): VGPR j, lanes 0-15 -> M=j, lanes 16-31 -> M=j+8.
template <int K, int NOUT>
__global__ void k_gemm_wmma(const float* __restrict__ A, const float* __restrict__ B,
                            float* __restrict__ C, int M) {
  constexpr int NT = NOUT / 16;
  const int lane  = threadIdx.x & 31;
  const int wave  = (int)((blockIdx.x * blockDim.x + threadIdx.x) >> 5);
  const int mtile = wave / NT;
  const int ntile = wave - mtile * NT;
  if (mtile * 16 >= M) return;                 // per-wave uniform: EXEC stays all-1s

  const int l15   = lane & 15;
  const int khalf = (lane >> 4) << 1;          // 0 for lanes 0-15, 2 for 16-31

  const float* __restrict__ ap = A + (size_t)(mtile * 16 + l15) * K + khalf;
  const float* __restrict__ bp = B + (size_t)khalf * NOUT + (ntile * 16 + l15);

  v8f acc = {};
#pragma unroll 8
  for (int k = 0; k < K; k += 4) {
    v2f a, b;
    a.x = ap[0];      a.y = ap[1];             // contiguous 8B per lane
    b.x = bp[0];      b.y = bp[NOUT];          // B is tiny (<=64KB) -> stays in cache
    acc = __builtin_amdgcn_wmma_f32_16x16x4_f32(
        /*neg_a=*/false, a, /*neg_b=*/false, b,
        /*c_mod=*/(short)0, acc, /*reuse_a=*/false, /*reuse_b=*/false);
    ap += 4;
    bp += 4 * NOUT;
  }

  const int rbase = mtile * 16 + ((lane >> 4) << 3);   // upper half-wave -> rows +8
  float* __restrict__ op = C + (size_t)rbase * NOUT + (ntile * 16 + l15);
#pragma unroll
  for (int j = 0; j < 8; ++j) op[(size_t)j * NOUT] = acc[j];
}

// ----------------- aggregation -----------------
// agg[i,f] = h[i,f] * dinv[i]^2 (self loop)  [+ bias for the final layer]
template <int F, bool ADD_BIAS>
__global__ void k_self_init(const float* __restrict__ h, const float* __restrict__ dinv,
                            const float* __restrict__ bias, float* __restrict__ agg, int n) {
  long long t = (long long)blockIdx.x * blockDim.x + threadIdx.x;
  if (t >= (long long)n * F) return;
  int i = (int)(t >> __builtin_ctz(F));
  int f = (int)(t & (F - 1));
  float di = dinv[i];
  float v  = h[t] * di * di;
  if (ADD_BIAS) v += bias[f];
  agg[t] = v;
}

// agg[dst,f] += h[src,f] * dinv[src]*w*dinv[dst]   (one thread per (edge, feature))
template <int F>
__global__ void k_scatter(const float* __restrict__ h, const float* __restrict__ dinv,
                          const int* __restrict__ src, const int* __restrict__ dst,
                          const float* __restrict__ w, float* __restrict__ agg, int e) {
  long long t = (long long)blockIdx.x * blockDim.x + threadIdx.x;
  if (t >= (long long)e * F) return;
  long long eidx = t >> __builtin_ctz(F);
  int f = (int)(t & (F - 1));
  int s = src[eidx];
  int d = dst[eidx];
  float nrm = dinv[s] * w[eidx] * dinv[d];
  unsafeAtomicAdd(&agg[(size_t)d * F + f], h[(size_t)s * F + f] * nrm);
}

template <int F>
__global__ void k_bias_relu(float* __restrict__ agg, const float* __restrict__ bias, int n) {
  long long t = (long long)blockIdx.x * blockDim.x + threadIdx.x;
  if (t >= (long long)n * F) return;
  int f = (int)(t & (F - 1));
  agg[t] = fmaxf(agg[t] + bias[f], 0.0f);
}

// ----------------- launch -----------------
extern "C" void kernel_launch(void* const* d_in, const int* in_sizes, int n_in,
                              void* d_out, int out_size, void* d_ws, size_t ws_size,
                              hipStream_t stream) {
  const float* x  = (const float*)d_in[0];
  const float* ew = (const float*)d_in[1];
  const float* W1 = (const float*)d_in[2];
  const float* b1 = (const float*)d_in[3];
  const float* W2 = (const float*)d_in[4];
  const float* b2 = (const float*)d_in[5];
  const int*   ei = (const int*)d_in[6];

  const int N = in_sizes[0] / KF;      // 100000 (multiple of 16)
  const int E = in_sizes[1];           // 1600000
  const int* src = ei;
  const int* dst = ei + E;

  // workspace layout: dinv[N] | h1[N*64] | agg1/h[N*64] | h2[N*32]  (~64.5 MB)
  char*  ws   = (char*)d_ws;
  float* dinv = (float*)ws;
  size_t off  = (((size_t)N * sizeof(float)) + 255) & ~(size_t)255;
  float* h1   = (float*)(ws + off);
  float* agg1 = h1 + (size_t)N * HD;
  float* h2   = agg1 + (size_t)N * HD;
  float* out  = (float*)d_out;

  const int B = 256;                   // 8 wave32s per block
  auto cdiv = [](long long a, long long b) { return (unsigned)((a + b - 1) / b); };

  // 1) degree (self-loop + edges) -> dinv = deg^{-1/2}
  k_deg_init <<<cdiv(N, B), B, 0, stream>>>(dinv, N);
  k_deg_edges<<<cdiv(E, B), B, 0, stream>>>(dst, ew, dinv, E);
  k_dinv     <<<cdiv(N, B), B, 0, stream>>>(dinv, N);

  // 2) h1 = x @ W1 (WMMA f32)
  {
    long long waves = (long long)((N + 15) / 16) * (HD / 16);
    k_gemm_wmma<KF, HD><<<cdiv(waves * 32, B), B, 0, stream>>>(x, W1, h1, N);
  }

  // 3) layer-1 aggregation: self loop, edge scatter, then bias + ReLU (in place)
  k_self_init<HD, false><<<cdiv((long long)N * HD, B), B, 0, stream>>>(h1, dinv, b1, agg1, N);
  k_scatter<HD>         <<<cdiv((long long)E * HD, B), B, 0, stream>>>(h1, dinv, src, dst, ew, agg1, E);
  k_bias_relu<HD>       <<<cdiv((long long)N * HD, B), B, 0, stream>>>(agg1, b1, N);

  // 4) h2 = h @ W2 (WMMA f32)
  {
    long long waves = (long long)((N + 15) / 16) * (CD / 16);
    k_gemm_wmma<HD, CD><<<cdiv(waves * 32, B), B, 0, stream>>>(agg1, W2, h2, N);
  }

  // 5) layer-2 aggregation straight into d_out (bias folded into self-loop init)
  k_self_init<CD, true><<<cdiv((long long)N * CD, B), B, 0, stream>>>(h2, dinv, b2, out, N);
  k_scatter<CD>        <<<cdiv((long long)E * CD, B), B, 0, stream>>>(h2, dinv, src, dst, ew, out, E);
}